// glp_linear_7146825581230
// MI455X (gfx1250) — compile-verified
//
#include <hip/hip_runtime.h>
#include <stdint.h>

typedef float v2f __attribute__((ext_vector_type(2)));
typedef float v8f __attribute__((ext_vector_type(8)));

#define M_TOTAL 1024
#define K_TOTAL 1024
#define N_TOTAL 128
#define M_BLK   64
#define KC      32
#define NKITER  (K_TOTAL / KC)        // 32

// LDS strides (in f32) padded for bank-conflict-free WMMA fragment reads,
// while keeping every async B128 store 16B-aligned (36*4=144B, 144*4=576B).
#define W_STRIDE 36
#define X_STRIDE 144
#define W_BUF_F32 (M_BLK * W_STRIDE)          // 2304
#define X_BUF_F32 (KC * X_STRIDE)             // 4608
#define BUF_F32   (W_BUF_F32 + X_BUF_F32)     // 6912
#define BUF_BYTES (BUF_F32 * 4)

// Async global->LDS copy, 16 bytes per lane. GV addressing (64-bit vaddr).
__device__ __forceinline__ void async_load_b128(uint32_t lds_addr, const float* gaddr) {
    asm volatile("global_load_async_to_lds_b128 %0, %1, off"
                 :: "v"(lds_addr), "v"(gaddr)
                 : "memory");
}

#define WAIT_ASYNCCNT(n) asm volatile("s_wait_asynccnt " #n ::: "memory")

__global__ void __launch_bounds__(256)
glp_linear_wmma_kernel(const float* __restrict__ X,   // [128, 1024, 128]
                       const float* __restrict__ W,   // [1024, 1024]
                       const float* __restrict__ bias,// [1024]
                       float* __restrict__ C)         // [128, 1024, 128]
{
    __shared__ float lds[2 * BUF_F32];   // 55296 bytes, double-buffered W+X tiles

    const int tid   = threadIdx.x;       // 0..255
    const int lane  = tid & 31;
    const int wave  = tid >> 5;          // 0..7
    const int lo    = lane & 15;
    const int hi    = lane >> 4;         // 0 or 1
    const int n0    = wave * 16;

    const int m_blk = blockIdx.x * M_BLK;
    const int batch = blockIdx.y;

    const float* Wg = W + (size_t)m_blk * K_TOTAL;
    const float* Xg = X + (size_t)batch * K_TOTAL * N_TOTAL;
    float*       Cg = C + (size_t)batch * M_TOTAL * N_TOTAL;

    const uint32_t lds_base = (uint32_t)(uintptr_t)(&lds[0]);

    // Issue one K-chunk (KC=32) of W (64x32) and X (32x128) into LDS buffer bi.
    // 6 async B128 loads per thread per chunk (2 for W, 4 for X).
    auto issue = [&](int k0, int bi) {
        const uint32_t wb = lds_base + (uint32_t)bi * BUF_BYTES;
        const uint32_t xb = wb + W_BUF_F32 * 4;
        #pragma unroll
        for (int j = 0; j < 2; ++j) {
            int q   = tid + j * 256;         // 0..511
            int row = q >> 3;                // 0..63
            int kq  = q & 7;                 // quad of 4 f32 along K
            async_load_b128(wb + (uint32_t)(row * W_STRIDE + kq * 4) * 4u,
                            Wg + (size_t)row * K_TOTAL + k0 + kq * 4);
        }
        #pragma unroll
        for (int j = 0; j < 4; ++j) {
            int q   = tid + j * 256;         // 0..1023
            int row = q >> 5;                // 0..31 (K within chunk)
            int nq  = q & 31;                // quad of 4 f32 along N
            async_load_b128(xb + (uint32_t)(row * X_STRIDE + nq * 4) * 4u,
                            Xg + (size_t)(k0 + row) * N_TOTAL + nq * 4);
        }
    };

    // Accumulators: 4 M-subtiles of 16x16, init from bias (vectorized: the 8
    // rows a lane holds are contiguous in bias -> one v8f load per subtile).
    // C/D layout: VGPR r holds row (mt*16 + 8*hi + r), lanes map to column lo.
    v8f acc[4];
    #pragma unroll
    for (int mt = 0; mt < 4; ++mt)
        acc[mt] = *(const v8f*)&bias[m_blk + mt * 16 + 8 * hi];

    issue(0, 0);   // prologue: fill buffer 0

    for (int kb = 0; kb < NKITER; ++kb) {
        WAIT_ASYNCCNT(0);     // this wave's loads for buf[kb&1] are in LDS
        __syncthreads();      // everyone's loads for buf[kb&1] are in LDS; and
                              // everyone is past compute(kb-1), so buf[(kb+1)&1]
                              // is safe to overwrite below.
        if (kb + 1 < NKITER)
            issue((kb + 1) * KC, (kb + 1) & 1);   // prefetch overlaps compute

        const float* wt = &lds[(kb & 1) * BUF_F32];
        const float* xt = wt + W_BUF_F32;

        #pragma unroll
        for (int kk = 0; kk < 8; ++kk) {            // 8 WMMA K-steps of 4
            const int krow = kk * 4 + 2 * hi;
            // B fragment (4x16): VGPR v -> K = v + 2*hi, lanes -> column lo
            v2f bf;
            bf.x = xt[(krow + 0) * X_STRIDE + n0 + lo];
            bf.y = xt[(krow + 1) * X_STRIDE + n0 + lo];
            #pragma unroll
            for (int mt = 0; mt < 4; ++mt) {
                // A fragment (16x4): lanes -> row lo, VGPR v -> K = v + 2*hi
                v2f af = *(const v2f*)&wt[(mt * 16 + lo) * W_STRIDE + krow];
                acc[mt] = __builtin_amdgcn_wmma_f32_16x16x4_f32(
                    /*neg_a=*/false, af, /*neg_b=*/false, bf,
                    /*c_mod=*/(short)0, acc[mt],
                    /*reuse_a=*/false, /*reuse_b=*/false);
            }
        }
    }

    // Store 64x128 tile: each half-wave writes 64B contiguous per (mt, r).
    #pragma unroll
    for (int mt = 0; mt < 4; ++mt) {
        #pragma unroll
        for (int r = 0; r < 8; ++r) {
            int row = m_blk + mt * 16 + 8 * hi + r;
            Cg[(size_t)row * N_TOTAL + n0 + lo] = acc[mt][r];
        }
    }
}

extern "C" void kernel_launch(void* const* d_in, const int* in_sizes, int n_in,
                              void* d_out, int out_size, void* d_ws, size_t ws_size,
                              hipStream_t stream) {
    const float* X    = (const float*)d_in[0];   // in_stack [128,1024,128]
    const float* W    = (const float*)d_in[1];   // [1024,1024]
    const float* bias = (const float*)d_in[2];   // [1024]
    float*       out  = (float*)d_out;           // [128,1024,128]

    dim3 grid(M_TOTAL / M_BLK, 128);             // 16 x 128
    dim3 block(256);
    glp_linear_wmma_kernel<<<grid, block, 0, stream>>>(X, W, bias, out);
}